// RGATLayer_9844065042606
// MI455X (gfx1250) — compile-verified
//
#include <hip/hip_runtime.h>
#include <hip/hip_bf16.h>

typedef float v2f __attribute__((ext_vector_type(2)));
typedef float v8f __attribute__((ext_vector_type(8)));

#define N_NODES 100000
#define N_EDGES 1600000
#define IN_DIM  128
#define OUT_DIM 128
#define NHEAD   4
#define HDIM    32
#define NEG_SLOPE 0.2f

// ---------- monotone float<->uint mapping so integer atomicMax == float max ----------
__device__ __forceinline__ unsigned floatFlip(float f) {
  unsigned u = __float_as_uint(f);
  unsigned mask = (unsigned)(-(int)(u >> 31)) | 0x80000000u;
  return u ^ mask;
}
__device__ __forceinline__ float floatUnflip(unsigned u) {
  unsigned mask = ((u >> 31) - 1u) | 0x80000000u;
  return __uint_as_float(u ^ mask);
}
#define FLIPPED_NEG_INF 0x007FFFFFu  // floatFlip(-inf)

// ---------------------------------------------------------------------------
// Kernel 1: feat[N,128] = h[N,128] @ W[128,128]^T  via V_WMMA_F32_16X16X4_F32
// One wave computes one 16(node)x16(out-feature) f32 tile; 8 waves/block cover
// all 128 output features of a 16-node stripe. K looped 128/4 = 32 WMMAs.
// A-layout (16x4 f32): lanes 0-15: v{0,1}=A[lane,K=0/1]; lanes 16-31: K=2/3.
// B-layout (4x16):     lanes 0-15: v{0,1}=B[K=0/1,lane]; lanes 16-31: K=2/3.
// D-layout (16x16):    vgpr r: lanes 0-15 -> M=r, lanes 16-31 -> M=r+8.
// ---------------------------------------------------------------------------
__global__ void __launch_bounds__(256)
gemm_feat_wmma(const float* __restrict__ h, const float* __restrict__ W,
               float* __restrict__ feat) {
  const int lane  = threadIdx.x & 31;
  const int wave  = threadIdx.x >> 5;      // 0..7 -> output-feature tile
  const int n0    = blockIdx.x * 16;       // node stripe (N divisible by 16)
  const int f0    = wave * 16;
  const int col   = lane & 15;             // M for A loads, N for B/D
  const int khalf = lane >> 4;             // 0 or 1

  const float* hrow = h + (size_t)(n0 + col) * IN_DIM;  // A: row = node
  const float* wrow = W + (size_t)(f0 + col) * IN_DIM;  // B: W[f, k] (transposed use)

  v8f c = {};
#pragma unroll 4
  for (int k0 = 0; k0 < IN_DIM; k0 += 4) {
    v2f a, b;
    a.x = hrow[k0 + 2 * khalf + 0];
    a.y = hrow[k0 + 2 * khalf + 1];
    b.x = wrow[k0 + 2 * khalf + 0];
    b.y = wrow[k0 + 2 * khalf + 1];
    // 8 args: (neg_a, A, neg_b, B, c_mod, C, reuse_a, reuse_b)
    c = __builtin_amdgcn_wmma_f32_16x16x4_f32(false, a, false, b, (short)0, c,
                                              false, false);
  }
  float* out = feat + (size_t)n0 * OUT_DIM + f0;
#pragma unroll
  for (int r = 0; r < 8; ++r)
    out[(size_t)(r + 8 * khalf) * OUT_DIM + col] = c[r];
}

// ---------------------------------------------------------------------------
// Kernel 2: per-(node,head) attention logits el/er; also init m (=-inf key)
// and denom (=0) since the workspace is poisoned.
// ---------------------------------------------------------------------------
__global__ void __launch_bounds__(256)
attn_logits(const float* __restrict__ feat, const float* __restrict__ attn_l,
            const float* __restrict__ attn_r, float* __restrict__ el,
            float* __restrict__ er, unsigned* __restrict__ mkeys,
            float* __restrict__ denom) {
  int i = blockIdx.x * blockDim.x + threadIdx.x;  // i = n*NHEAD + head
  if (i >= N_NODES * NHEAD) return;
  int hh = i & (NHEAD - 1);
  const float* f  = feat + (size_t)(i >> 2) * OUT_DIM + hh * HDIM;
  const float* al = attn_l + hh * HDIM;
  const float* ar = attn_r + hh * HDIM;
  float sl = 0.f, sr = 0.f;
#pragma unroll
  for (int d = 0; d < HDIM; ++d) {
    float v = f[d];
    sl += v * al[d];
    sr += v * ar[d];
  }
  el[i] = sl;
  er[i] = sr;
  mkeys[i] = FLIPPED_NEG_INF;
  denom[i] = 0.f;
}

// ---------------------------------------------------------------------------
// Kernel 3: init d_out with mean-over-heads of bias (segment sums land on top)
// ---------------------------------------------------------------------------
__global__ void __launch_bounds__(256)
init_out(const float* __restrict__ bias, float* __restrict__ out) {
  int i = blockIdx.x * blockDim.x + threadIdx.x;  // i = n*HDIM + d
  if (i >= N_NODES * HDIM) return;
  int d = i & (HDIM - 1);
  out[i] = 0.25f * (bias[d] + bias[HDIM + d] + bias[2 * HDIM + d] + bias[3 * HDIM + d]);
}

// ---------------------------------------------------------------------------
// Kernel 4: segment max of leaky-relu edge scores (int-key float atomic max)
// ---------------------------------------------------------------------------
__global__ void __launch_bounds__(256)
edge_max(const int* __restrict__ src, const int* __restrict__ dst,
         const float* __restrict__ el, const float* __restrict__ er,
         unsigned* __restrict__ mkeys) {
  int e = blockIdx.x * blockDim.x + threadIdx.x;
  if (e >= N_EDGES) return;
  int s = src[e], d = dst[e];
#pragma unroll
  for (int hh = 0; hh < NHEAD; ++hh) {
    float v = el[s * NHEAD + hh] + er[d * NHEAD + hh];
    v = v > 0.f ? v : NEG_SLOPE * v;
    atomicMax(&mkeys[d * NHEAD + hh], floatFlip(v));
  }
}

// ---------------------------------------------------------------------------
// Kernel 5: segment sum of exp(e - m)  (softmax denominator)
// ---------------------------------------------------------------------------
__global__ void __launch_bounds__(256)
edge_denom(const int* __restrict__ src, const int* __restrict__ dst,
           const float* __restrict__ el, const float* __restrict__ er,
           const unsigned* __restrict__ mkeys, float* __restrict__ denom) {
  int e = blockIdx.x * blockDim.x + threadIdx.x;
  if (e >= N_EDGES) return;
  int s = src[e], d = dst[e];
#pragma unroll
  for (int hh = 0; hh < NHEAD; ++hh) {
    float v = el[s * NHEAD + hh] + er[d * NHEAD + hh];
    v = v > 0.f ? v : NEG_SLOPE * v;
    float a = __expf(v - floatUnflip(mkeys[d * NHEAD + hh]));
    unsafeAtomicAdd(&denom[d * NHEAD + hh], a);
  }
}

// ---------------------------------------------------------------------------
// Kernel 6: weighted scatter-aggregate, head-mean folded into the accumulate:
//   out[dst, d] += 0.25 * sum_h alpha[e,h] * feat[src, h*32 + d]
// One wave per edge; lane = per-head feature dim d (coalesced 128B per head).
// ---------------------------------------------------------------------------
__global__ void __launch_bounds__(256)
edge_aggregate(const int* __restrict__ src, const int* __restrict__ dst,
               const float* __restrict__ el, const float* __restrict__ er,
               const unsigned* __restrict__ mkeys, const float* __restrict__ denom,
               const float* __restrict__ feat, float* __restrict__ out) {
  int eid = blockIdx.x * 8 + (threadIdx.x >> 5);
  if (eid >= N_EDGES) return;
  int lane = threadIdx.x & 31;
  int s = src[eid], d = dst[eid];

  float alpha = 0.f;
  if (lane < NHEAD) {
    float v = el[s * NHEAD + lane] + er[d * NHEAD + lane];
    v = v > 0.f ? v : NEG_SLOPE * v;
    alpha = __expf(v - floatUnflip(mkeys[d * NHEAD + lane])) /
            denom[d * NHEAD + lane];
  }
  float a0 = __shfl(alpha, 0, 32);
  float a1 = __shfl(alpha, 1, 32);
  float a2 = __shfl(alpha, 2, 32);
  float a3 = __shfl(alpha, 3, 32);

  const float* f = feat + (size_t)s * OUT_DIM;
  float val = 0.25f * (a0 * f[lane] + a1 * f[HDIM + lane] +
                       a2 * f[2 * HDIM + lane] + a3 * f[3 * HDIM + lane]);
  unsafeAtomicAdd(&out[(size_t)d * HDIM + lane], val);
}

// ---------------------------------------------------------------------------
extern "C" void kernel_launch(void* const* d_in, const int* in_sizes, int n_in,
                              void* d_out, int out_size, void* d_ws, size_t ws_size,
                              hipStream_t stream) {
  const float* h       = (const float*)d_in[0];
  const int*   src     = (const int*)d_in[1];
  const int*   dst     = (const int*)d_in[2];
  // d_in[3] = rel_types, d_in[8] = rel_emb : dead inputs (never read by GATConv)
  const float* W       = (const float*)d_in[4];
  const float* attn_l  = (const float*)d_in[5];
  const float* attn_r  = (const float*)d_in[6];
  const float* bias    = (const float*)d_in[7];
  float* out = (float*)d_out;

  // workspace layout (floats): feat | el | er | mkeys | denom  (~57.6 MB)
  float*    feat  = (float*)d_ws;
  float*    el    = feat + (size_t)N_NODES * OUT_DIM;
  float*    er    = el + (size_t)N_NODES * NHEAD;
  unsigned* mkeys = (unsigned*)(er + (size_t)N_NODES * NHEAD);
  float*    denom = (float*)(mkeys + (size_t)N_NODES * NHEAD);

  // 1) feat = h @ W^T  (WMMA f32)
  gemm_feat_wmma<<<N_NODES / 16, 256, 0, stream>>>(h, W, feat);
  // 2) el/er + init m, denom
  attn_logits<<<(N_NODES * NHEAD + 255) / 256, 256, 0, stream>>>(
      feat, attn_l, attn_r, el, er, mkeys, denom);
  // 3) init output with head-mean of bias
  init_out<<<(N_NODES * HDIM + 255) / 256, 256, 0, stream>>>(bias, out);
  // 4) segment max
  edge_max<<<(N_EDGES + 255) / 256, 256, 0, stream>>>(src, dst, el, er, mkeys);
  // 5) softmax denominator
  edge_denom<<<(N_EDGES + 255) / 256, 256, 0, stream>>>(src, dst, el, er, mkeys, denom);
  // 6) weighted aggregate with folded head-mean
  edge_aggregate<<<(N_EDGES + 7) / 8, 256, 0, stream>>>(src, dst, el, er, mkeys,
                                                        denom, feat, out);
}